// KGCL_22333829939491
// MI455X (gfx1250) — compile-verified
//
#include <hip/hip_runtime.h>
#include <hip/hip_bf16.h>
#include <math.h>

#define N_ENT  100000
#define N_RELS 20
#define N_EDGE 1000000
#define CDIM   64

typedef __attribute__((ext_vector_type(2))) float v2f;
typedef __attribute__((ext_vector_type(8))) float v8f;

// ---- monotonic float<->int mapping so atomicMax(int) implements float max ----
__device__ __forceinline__ int f2ord(float f) {
  int i = __float_as_int(f);
  return (i >= 0) ? i : (i ^ 0x7fffffff);
}
__device__ __forceinline__ float ord2f(int i) {
  return __int_as_float((i >= 0) ? i : (i ^ 0x7fffffff));
}

// ---------------------------------------------------------------------------
// Kernel 0 (once): fold fc_w into per-relation head/tail vectors.
// Wmat is K-major [64][48]: col j<20 -> wh_j[k], 20<=j<40 -> wt_{j-20}[k], pad 0.
// brel[r] = rel[r] . fc_b
// ---------------------------------------------------------------------------
__global__ void kg_precompute_w(const float* __restrict__ rel,
                                const float* __restrict__ fcw,   // [64][128]
                                const float* __restrict__ fcb,   // [64]
                                float* __restrict__ Wmat,        // [64][48]
                                float* __restrict__ brel) {      // [20]
  int gid = blockIdx.x * blockDim.x + threadIdx.x;
  if (gid < 64 * 48) {
    int k = gid / 48, j = gid % 48;
    float v = 0.f;
    if (j < 40) {
      int r = (j < 20) ? j : (j - 20);
      int i = (j < 20) ? k : (64 + k);
      const float* rr = rel + r * 64;
      #pragma unroll 8
      for (int o = 0; o < 64; ++o) v += rr[o] * fcw[o * 128 + i];
    }
    Wmat[k * 48 + j] = v;
  } else if (gid < 64 * 48 + N_RELS) {
    int r = gid - 64 * 48;
    float v = 0.f;
    const float* rr = rel + r * 64;
    for (int o = 0; o < 64; ++o) v += rr[o] * fcb[o];
    brel[r] = v;
  }
}

// ---------------------------------------------------------------------------
// Kernel 1: score GEMM via WMMA f32 16x16x4.
// [N_ENT x 64] @ [64 x 48] -> H[N_ENT][20], T[N_ENT][20] (cols 40..47 dropped)
// One wave per 16-entity M-tile; 3 N-tiles of 16; K = 16 steps of 4.
// LDS holds W pair-interleaved: lw[(k>>1)*48 + col] is the v2f {W[k],W[k+1]}
// so each B fragment is a single aligned ds_load_b64.
// ---------------------------------------------------------------------------
__global__ __launch_bounds__(256) void kg_score_wmma(
    const float* __restrict__ ent, const float* __restrict__ Wmat,
    float* __restrict__ H, float* __restrict__ T) {
  __shared__ v2f lw[32 * 48];
  {
    float* lwf = (float*)lw;
    for (int i = threadIdx.x; i < 64 * 48; i += 256) {
      int k = i / 48, j = i % 48;
      lwf[((k >> 1) * 48 + j) * 2 + (k & 1)] = Wmat[i];
    }
  }
  __syncthreads();

  int wave = threadIdx.x >> 5;
  int lane = threadIdx.x & 31;
  int half = lane >> 4;    // 0: lanes 0-15, 1: lanes 16-31
  int lid  = lane & 15;
  int m0 = (blockIdx.x * 8 + wave) * 16;
  if (m0 >= N_ENT) return;   // wave-uniform -> EXEC all 1s inside

  v8f acc0 = {}; v8f acc1 = {}; v8f acc2 = {};
  const float* arow = ent + (size_t)(m0 + lid) * CDIM;

  #pragma unroll 4
  for (int kk = 0; kk < 16; ++kk) {
    int kb = kk * 4 + 2 * half;                 // even
    // A 16x4 f32: lanes0-15 {K0,K1}, lanes16-31 {K2,K3}, row = lid
    v2f a = *(const v2f*)(arow + kb);
    // B 4x16 f32: lanes0-15 {K0,K1} at N=lid, lanes16-31 {K2,K3}
    int base = (kk * 2 + half) * 48;
    v2f b0 = lw[base + lid];
    v2f b1 = lw[base + 16 + lid];
    v2f b2 = lw[base + 32 + lid];
    acc0 = __builtin_amdgcn_wmma_f32_16x16x4_f32(false, a, false, b0, (short)0, acc0, false, false);
    acc1 = __builtin_amdgcn_wmma_f32_16x16x4_f32(false, a, false, b1, (short)0, acc1, false, false);
    acc2 = __builtin_amdgcn_wmma_f32_16x16x4_f32(false, a, false, b2, (short)0, acc2, false, false);
  }

  // C/D layout: lanes0-15: VGPR v -> row m0+v, col lid; lanes16-31: row m0+8+v
  int rbase = m0 + half * 8;
  #pragma unroll
  for (int v = 0; v < 8; ++v) {
    int row = rbase + v;
    H[row * 20 + lid] = acc0[v];                       // cols 0..15
    int c1 = 16 + lid;                                 // cols 16..31
    if (c1 < 20) H[row * 20 + c1] = acc1[v];
    else         T[row * 20 + (c1 - 20)] = acc1[v];
    int c2 = 32 + lid;                                 // cols 32..47
    if (c2 < 40) T[row * 20 + (c2 - 20)] = acc2[v];
  }
}

// ---------------------------------------------------------------------------
// Kernel 2: per-hop init: out = ent (residual), segment max = -inf, denom = 0
// ---------------------------------------------------------------------------
__global__ void kg_init(const float4* __restrict__ ein4, float4* __restrict__ eout4,
                        int* __restrict__ mOrd, float* __restrict__ denom) {
  int gid = blockIdx.x * blockDim.x + threadIdx.x;
  if (gid < N_ENT * (CDIM / 4)) eout4[gid] = ein4[gid];
  if (gid < N_ENT) { mOrd[gid] = (int)0x80000000; denom[gid] = 0.f; }
}

// ---------------------------------------------------------------------------
// Kernel 3: edge scores + segment max over head
// ---------------------------------------------------------------------------
__global__ void kg_edge_score(const float* __restrict__ H, const float* __restrict__ T,
                              const float* __restrict__ brel,
                              const int* __restrict__ head, const int* __restrict__ tail,
                              const int* __restrict__ etype,
                              float* __restrict__ eArr, int* __restrict__ mOrd) {
  int idx = blockIdx.x * blockDim.x + threadIdx.x;
  if (idx >= N_EDGE) return;
  int h = head[idx], t = tail[idx], r = etype[idx];
  float e = H[h * 20 + r] + T[t * 20 + r] + brel[r];
  e = (e > 0.f) ? e : 0.2f * e;            // leaky_relu(0.2)
  eArr[idx] = e;
  atomicMax(&mOrd[h], f2ord(e));
}

// ---------------------------------------------------------------------------
// Kernel 4: exp(e - m[head]) + segment sum into denom
// ---------------------------------------------------------------------------
__global__ void kg_edge_exp(float* __restrict__ eArr, const int* __restrict__ mOrd,
                            const int* __restrict__ head, float* __restrict__ denom) {
  int idx = blockIdx.x * blockDim.x + threadIdx.x;
  if (idx >= N_EDGE) return;
  int h = head[idx];
  float x = expf(eArr[idx] - ord2f(mOrd[h]));
  eArr[idx] = x;
  atomicAdd(&denom[h], x);
}

// ---------------------------------------------------------------------------
// Kernel 5: weighted scatter-sum: out[head] += alpha * ent[tail]
// one thread per (edge, 4-channel group): float4 gather + 4 float atomics;
// all tables are L2-resident
// ---------------------------------------------------------------------------
__global__ void kg_aggregate(const float* __restrict__ eArr, const float* __restrict__ denom,
                             const int* __restrict__ head, const int* __restrict__ tail,
                             const float* __restrict__ ein, float* __restrict__ eout) {
  int gid = blockIdx.x * blockDim.x + threadIdx.x;
  if (gid >= N_EDGE * (CDIM / 4)) return;
  int edge = gid >> 4;
  int c = (gid & 15) << 2;
  int h = head[edge], t = tail[edge];
  float alpha = eArr[edge] / denom[h];
  float4 v = *(const float4*)(ein + (size_t)t * CDIM + c);
  float* dst = eout + (size_t)h * CDIM + c;
  atomicAdd(dst + 0, alpha * v.x);
  atomicAdd(dst + 1, alpha * v.y);
  atomicAdd(dst + 2, alpha * v.z);
  atomicAdd(dst + 3, alpha * v.w);
}

// ---------------------------------------------------------------------------
// Kernel 6: row L2-normalize, one wave32 per entity; contiguous float2/lane
// ---------------------------------------------------------------------------
__global__ void kg_normalize(float* __restrict__ ent) {
  int gid = blockIdx.x * blockDim.x + threadIdx.x;
  int row = gid >> 5, lane = gid & 31;
  if (row >= N_ENT) return;
  float* p = ent + (size_t)row * CDIM + lane * 2;
  v2f v = *(const v2f*)p;
  float s = v.x * v.x + v.y * v.y;
  #pragma unroll
  for (int off = 16; off > 0; off >>= 1) s += __shfl_xor(s, off, 32);
  float inv = 1.f / fmaxf(sqrtf(s), 1e-12f);
  v.x *= inv; v.y *= inv;
  *(v2f*)p = v;
}

// ---------------------------------------------------------------------------
static inline size_t alignup(size_t x) { return (x + 255) & ~(size_t)255; }

extern "C" void kernel_launch(void* const* d_in, const int* in_sizes, int n_in,
                              void* d_out, int out_size, void* d_ws, size_t ws_size,
                              hipStream_t stream) {
  const float* ent0 = (const float*)d_in[0];   // [100000,64]
  const float* rel  = (const float*)d_in[1];   // [20,64]
  const int*   eidx = (const int*)d_in[2];     // [2,1e6]
  const int*   ety  = (const int*)d_in[3];     // [1e6]
  const float* fcw  = (const float*)d_in[4];   // [64,128]
  const float* fcb  = (const float*)d_in[5];   // [64]
  float* out = (float*)d_out;

  const int* head = eidx;
  const int* tail = eidx + N_EDGE;

  // workspace carve-out
  char* ws = (char*)d_ws;
  size_t off = 0;
  float* Wmat = (float*)(ws + off); off = alignup(off + 64 * 48 * sizeof(float));
  float* brel = (float*)(ws + off); off = alignup(off + N_RELS * sizeof(float));
  float* H    = (float*)(ws + off); off = alignup(off + (size_t)N_ENT * 20 * sizeof(float));
  float* T    = (float*)(ws + off); off = alignup(off + (size_t)N_ENT * 20 * sizeof(float));
  float* eArr = (float*)(ws + off); off = alignup(off + (size_t)N_EDGE * sizeof(float));
  int*   mOrd = (int*)  (ws + off); off = alignup(off + (size_t)N_ENT * sizeof(int));
  float* den  = (float*)(ws + off); off = alignup(off + (size_t)N_ENT * sizeof(float));
  float* ent1 = (float*)(ws + off); off = alignup(off + (size_t)N_ENT * CDIM * sizeof(float));
  (void)ws_size; (void)in_sizes; (void)n_in; (void)out_size;

  // fold fc_w into per-relation vectors (shared by both hops)
  kg_precompute_w<<<(64 * 48 + N_RELS + 255) / 256, 256, 0, stream>>>(rel, fcw, fcb, Wmat, brel);

  const int scoreBlocks = (N_ENT / 16 + 7) / 8;                  // 782
  const int initBlocks  = (N_ENT * (CDIM / 4) + 255) / 256;      // 6250
  const int edgeBlocks  = (N_EDGE + 255) / 256;                  // 3907
  const int aggBlocks   = (N_EDGE * (CDIM / 4)) / 256;           // 62500
  const int normBlocks  = (N_ENT * 32 + 255) / 256;              // 12500

  for (int hop = 0; hop < 2; ++hop) {
    const float* ein = (hop == 0) ? ent0 : ent1;
    float* eout      = (hop == 0) ? ent1 : out;

    kg_score_wmma<<<scoreBlocks, 256, 0, stream>>>(ein, Wmat, H, T);
    kg_init<<<initBlocks, 256, 0, stream>>>((const float4*)ein, (float4*)eout, mOrd, den);
    kg_edge_score<<<edgeBlocks, 256, 0, stream>>>(H, T, brel, head, tail, ety, eArr, mOrd);
    kg_edge_exp<<<edgeBlocks, 256, 0, stream>>>(eArr, mOrd, head, den);
    kg_aggregate<<<aggBlocks, 256, 0, stream>>>(eArr, den, head, tail, ein, eout);
    kg_normalize<<<normBlocks, 256, 0, stream>>>(eout);
  }
}